// neigh_Conv2_30434138259617
// MI455X (gfx1250) — compile-verified
//
#include <hip/hip_runtime.h>
#include <hip/hip_bf16.h>

typedef __attribute__((ext_vector_type(2))) float v2f;
typedef __attribute__((ext_vector_type(8))) float v8f;

// out[b,c,p] = x[b,c,p] + sum_k W[c,k] * x[b, start(c)+k, p] + bias[c]
// start(c) = clip(c-2, 0, C-K).  Cast as banded GEMM on V_WMMA_F32_16X16X4_F32.
// Each wave: one 16-channel tile x 64 pixels (4 WMMA tiles), 5 chained WMMAs per tile.
__global__ __launch_bounds__(256) void neigh_conv_wmma_kernel(
    const float* __restrict__ x,     // [B, C, 4096]
    const float* __restrict__ W,     // [C, 5]
    const float* __restrict__ bias,  // [C]
    float* __restrict__ out)         // [B, C, 4096]
{
    constexpr int C   = 200;
    constexpr int K   = 5;
    constexpr int P   = 4096;        // HW*HW
    constexpr int CK  = C - K;       // 195 (clip upper bound)
    constexpr int NPT = 4;           // pixel tiles per wave

    const int lane = threadIdx.x & 31;
    const int wave = threadIdx.x >> 5;
    const int m    = lane & 15;      // A: row M; B/C/D: pixel column N
    const int h    = lane >> 4;      // lane-half select per ISA VGPR layouts

    const int cm    = blockIdx.y * 16;                         // channel-tile base
    const int batch = blockIdx.z;
    const int p0    = (blockIdx.x * 8 + wave) * (NPT * 16);    // 64-pixel span base

    const float* xb = x   + (size_t)batch * C * P;
    float*       ob = out + (size_t)batch * C * P;

    // ---- Build A fragments (16x4 f32 per step; 5 steps cover channels cm-2..cm+17).
    // Lane owns output row c = cm + m.  A VGPR0 col K=2h, VGPR1 col K=1+2h.
    const int c      = cm + m;
    const int cc     = (c < C) ? c : (C - 1);
    const int startc = min(max(cc - 2, 0), CK);
    const int cbase  = cm - 2;

    v2f a[5];
#pragma unroll
    for (int s = 0; s < 5; ++s) {
        const int ch0 = cbase + 4 * s + 2 * h;
        const int ch1 = ch0 + 1;
        const int d0  = ch0 - startc;
        const int d1  = ch1 - startc;
        const bool in0 = (c < C) && (d0 >= 0) && (d0 < K);
        const bool in1 = (c < C) && (d1 >= 0) && (d1 < K);
        const float w0 = W[cc * K + min(max(d0, 0), K - 1)];   // safe addr; selected
        const float w1 = W[cc * K + min(max(d1, 0), K - 1)];
        a[s][0] = in0 ? w0 : 0.0f;
        a[s][1] = in1 ? w1 : 0.0f;
    }

    // ---- Band-row base pointers (clamped; clamped rows have all-zero A columns).
    const float* brow0[5];
    const float* brow1[5];
#pragma unroll
    for (int s = 0; s < 5; ++s) {
        const int chb = cbase + 4 * s + 2 * h;
        brow0[s] = xb + (size_t)min(max(chb,     0), C - 1) * P;
        brow1[s] = xb + (size_t)min(max(chb + 1, 0), C - 1) * P;
    }

    // ---- Bias accumulator template: D VGPR v holds row cm + v + 8h, col m.
    // Row validity is uniform over v (C % 16 == 8): only cm==192,h==1 is invalid.
    const int  row0     = cm + 8 * h;
    const bool rowvalid = (row0 + 7) < C;
    v8f accb;
#pragma unroll
    for (int v = 0; v < 8; ++v) {
        const int row = min(row0 + v, C - 1);
        accb[v] = bias[row];                    // junk ok for invalid rows (never stored)
    }

    // ---- 4 pixel tiles: 5 chained f32 WMMAs each, then residual add + store.
#pragma unroll
    for (int t = 0; t < NPT; ++t) {
        const int p = p0 + t * 16 + m;
        v8f acc = accb;
#pragma unroll
        for (int s = 0; s < 5; ++s) {
            v2f bfrag;
            bfrag[0] = brow0[s][p];             // B row K=2h
            bfrag[1] = brow1[s][p];             // B row K=1+2h
            acc = __builtin_amdgcn_wmma_f32_16x16x4_f32(
                /*neg_a=*/false, a[s], /*neg_b=*/false, bfrag,
                /*c_mod=*/(short)0, acc, /*reuse_a=*/false, /*reuse_b=*/false);
        }
        if (rowvalid) {                          // single uniform-per-lane predicate
#pragma unroll
            for (int v = 0; v < 8; ++v) {
                const size_t off = (size_t)(row0 + v) * P + p;
                ob[off] = xb[off] + acc[v];
            }
        }
    }
}

extern "C" void kernel_launch(void* const* d_in, const int* in_sizes, int n_in,
                              void* d_out, int out_size, void* d_ws, size_t ws_size,
                              hipStream_t stream) {
    const float* x    = (const float*)d_in[0];  // (16, 200, 64, 64) f32
    const float* W    = (const float*)d_in[1];  // (200, 5) f32
    const float* bias = (const float*)d_in[2];  // (200,) f32
    float* out = (float*)d_out;

    // 4096 px / (8 waves * 4 tiles * 16 px) = 8 pixel blocks; 13 channel tiles; 16 batches.
    dim3 grid(8, 13, 16);
    dim3 block(256, 1, 1);
    neigh_conv_wmma_kernel<<<grid, block, 0, stream>>>(x, W, bias, out);
}